// TtMambaSSM_84653805404554
// MI455X (gfx1250) — compile-verified
//
#include <hip/hip_runtime.h>
#include <hip/hip_bf16.h>
#include <math.h>

// Problem dims (match reference)
#define BATCH 128
#define E     5120
#define R     160
#define NPAD  32     // padded d_state
#define NS    16     // real d_state (cols >= NS contribute exactly 0 to output)

typedef float v2f __attribute__((ext_vector_type(2)));
typedef float v8f __attribute__((ext_vector_type(8)));

// Full-precision CDNA5 WMMA: D(16x16 f32) = A(16x4 f32) * B(4x16 f32) + C
// 8-arg form: (neg_a, A, neg_b, B, c_mod, C, reuse_a, reuse_b)
__device__ __forceinline__ v8f wmma4_f32(v2f a, v2f b, v8f c) {
  return __builtin_amdgcn_wmma_f32_16x16x4_f32(false, a, false, b, (short)0, c,
                                               false, false);
}

// ---------------------------------------------------------------------------
// Kernel 1: T[128,160] = x[128,5120] @ W_dt_low[5120,160]
// One wave32 per 16x16 output tile; K-loop over E in steps of 4.
// A-layout: lane(half,l): a = { x[m=l][k+2h], x[m=l][k+2h+1] }
// B-layout: lane(half,l): b = { W[k+2h][n=l], W[k+2h+1][n=l] }
// ---------------------------------------------------------------------------
__global__ __launch_bounds__(64) void k_gemm_xWlow(
    const float* __restrict__ x, const float* __restrict__ Wlow,
    float* __restrict__ T)
{
  const int lane  = threadIdx.x;
  const int half  = lane >> 4;
  const int l     = lane & 15;
  const int nTile = (blockIdx.x * blockDim.y + threadIdx.y) * 16;  // over R
  const int mTile = blockIdx.y * 16;                               // over BATCH

  v8f acc = {};
  const float* ap0 = x    + (size_t)(mTile + l) * E + 2 * half;
  const float* bp0 = Wlow + (size_t)(2 * half) * R + nTile + l;
  for (int k = 0; k < E; k += 4) {
    v2f a = *(const v2f*)(ap0 + k);                 // 8B-aligned (k even)
    v2f b;
    b.x = bp0[(size_t)k * R];
    b.y = bp0[(size_t)(k + 1) * R];
    acc = wmma4_f32(a, b, acc);
  }
  #pragma unroll
  for (int r = 0; r < 8; ++r)
    T[(size_t)(mTile + r + 8 * half) * R + nTile + l] = acc[r];
}

// ---------------------------------------------------------------------------
// Kernel 2: delta[128,5120] = softplus(T[128,160] @ W_dt[160,5120] + b_dt)
// Fused bias + thresholded softplus on the WMMA accumulators.
// ---------------------------------------------------------------------------
__global__ __launch_bounds__(128) void k_gemm_dt(
    const float* __restrict__ T, const float* __restrict__ Wdt,
    const float* __restrict__ bdt, float* __restrict__ delta)
{
  const int lane  = threadIdx.x;
  const int half  = lane >> 4;
  const int l     = lane & 15;
  const int nTile = (blockIdx.x * blockDim.y + threadIdx.y) * 16;  // over E
  const int mTile = blockIdx.y * 16;                               // over BATCH

  v8f acc = {};
  const float* ap0 = T   + (size_t)(mTile + l) * R + 2 * half;
  const float* bp0 = Wdt + (size_t)(2 * half) * E + nTile + l;
  for (int k = 0; k < R; k += 4) {
    v2f a = *(const v2f*)(ap0 + k);
    v2f b;
    b.x = bp0[(size_t)k * E];
    b.y = bp0[(size_t)(k + 1) * E];
    acc = wmma4_f32(a, b, acc);
  }
  const float bias = bdt[nTile + l];
  #pragma unroll
  for (int r = 0; r < 8; ++r) {
    float z  = acc[r] + bias;
    float sp = (z > 20.0f) ? z : log1pf(__expf(z));   // softplus, thresh=20
    delta[(size_t)(mTile + r + 8 * half) * E + nTile + l] = sp;
  }
}

// ---------------------------------------------------------------------------
// Kernel 3: Bp[128,16] = x @ W_B[:, :16],  Cp[128,16] = x @ W_C[:, :16]
// Columns >= NS of W_B/W_C are exactly zero in the reference -> skip them.
// threadIdx.y selects B vs C (wave-uniform, EXEC stays all-ones).
// ---------------------------------------------------------------------------
__global__ __launch_bounds__(64) void k_gemm_bc(
    const float* __restrict__ x, const float* __restrict__ WB,
    const float* __restrict__ WC, float* __restrict__ Bp,
    float* __restrict__ Cp)
{
  const int lane  = threadIdx.x;
  const int half  = lane >> 4;
  const int l     = lane & 15;
  const int sel   = threadIdx.y;          // 0 -> B, 1 -> C
  const int mTile = blockIdx.x * 16;      // over BATCH

  const float* W = sel ? WC : WB;
  float*       P = sel ? Cp : Bp;

  v8f acc = {};
  const float* ap0 = x + (size_t)(mTile + l) * E + 2 * half;
  const float* bp0 = W + (size_t)(2 * half) * NPAD + l;
  for (int k = 0; k < E; k += 4) {
    v2f a = *(const v2f*)(ap0 + k);
    v2f b;
    b.x = bp0[(size_t)k * NPAD];
    b.y = bp0[(size_t)(k + 1) * NPAD];
    acc = wmma4_f32(a, b, acc);
  }
  #pragma unroll
  for (int r = 0; r < 8; ++r)
    P[(size_t)(mTile + r + 8 * half) * NS + l] = acc[r];
}

// ---------------------------------------------------------------------------
// Kernel 4: memory-bound SSM update.
//   y[b,e]   = sum_{n<16} ( exp(delta*A[e,n]) * h[b,e,n] + delta*x*Bp[b,n] ) * Cp[b,n]
//   out[b,e] = x[b,e] * D[e] + y[b,e]
// One thread per (b,e): 4x float4 loads of h (first 64B of the 128B row only),
// A row, Bp/Cp rows; 16 v_exp_f32; 1 store. ~50 MB total HBM traffic.
// ---------------------------------------------------------------------------
__global__ __launch_bounds__(256) void k_ssm_update(
    const float* __restrict__ x, const float* __restrict__ h,
    const float* __restrict__ delta, const float* __restrict__ Bp,
    const float* __restrict__ Cp, const float* __restrict__ A,
    const float* __restrict__ Dvec, float* __restrict__ out)
{
  const int idx = blockIdx.x * 256 + threadIdx.x;   // over BATCH*E
  if (idx >= BATCH * E) return;
  const int b = idx / E;
  const int e = idx - b * E;

  const float xv = x[idx];
  const float d  = delta[idx];
  const float dx = d * xv;

  const float4* hp = (const float4*)(h + (size_t)idx * NPAD);  // h[b,e,0:16]
  const float4* Ap = (const float4*)(A + (size_t)e * NPAD);    // A[e,0:16]
  const float4* Bq = (const float4*)(Bp + (size_t)b * NS);
  const float4* Cq = (const float4*)(Cp + (size_t)b * NS);

  float y = 0.0f;
  #pragma unroll
  for (int q = 0; q < 4; ++q) {
    const float4 hv = hp[q], Av = Ap[q], Bv = Bq[q], Cv = Cq[q];
    y += (__expf(d * Av.x) * hv.x + dx * Bv.x) * Cv.x;
    y += (__expf(d * Av.y) * hv.y + dx * Bv.y) * Cv.y;
    y += (__expf(d * Av.z) * hv.z + dx * Bv.z) * Cv.z;
    y += (__expf(d * Av.w) * hv.w + dx * Bv.w) * Cv.w;
  }
  out[idx] = xv * Dvec[e] + y;
}

// ---------------------------------------------------------------------------
extern "C" void kernel_launch(void* const* d_in, const int* in_sizes, int n_in,
                              void* d_out, int out_size, void* d_ws,
                              size_t ws_size, hipStream_t stream)
{
  (void)in_sizes; (void)n_in; (void)out_size; (void)ws_size;
  const float* x    = (const float*)d_in[0];
  const float* h    = (const float*)d_in[1];
  const float* Wlow = (const float*)d_in[2];
  const float* Wdt  = (const float*)d_in[3];
  const float* bdt  = (const float*)d_in[4];
  const float* WB   = (const float*)d_in[5];
  const float* WC   = (const float*)d_in[6];
  const float* A    = (const float*)d_in[7];
  const float* Dv   = (const float*)d_in[8];
  float* out = (float*)d_out;

  // Workspace layout (floats): T | delta | Bp | Cp  (~2.7 MB total)
  float* T     = (float*)d_ws;
  float* delta = T + BATCH * R;
  float* Bp    = delta + (size_t)BATCH * E;
  float* Cp    = Bp + BATCH * NS;

  // 1) T = x @ W_dt_low          (10 n-tiles over R, 8 m-tiles over BATCH)
  k_gemm_xWlow<<<dim3(5, 8), dim3(32, 2), 0, stream>>>(x, Wlow, T);
  // 2) delta = softplus(T @ W_dt + b)   (320 n-tiles over E)
  k_gemm_dt<<<dim3(80, 8), dim3(32, 4), 0, stream>>>(T, Wdt, bdt, delta);
  // 3) Bp/Cp = x @ W_B/W_C (first 16 cols only)
  k_gemm_bc<<<dim3(8), dim3(32, 2), 0, stream>>>(x, WB, WC, Bp, Cp);
  // 4) fused SSM state update + C-contraction + skip connection
  k_ssm_update<<<dim3((BATCH * E + 255) / 256), dim3(256), 0, stream>>>(
      x, h, delta, Bp, Cp, A, Dv, out);
}